// DyBEMLayer_893353198381
// MI455X (gfx1250) — compile-verified
//
#include <hip/hip_runtime.h>
#include <hip/hip_bf16.h>

typedef float v2f __attribute__((ext_vector_type(2)));
typedef float v8f __attribute__((ext_vector_type(8)));

#define IN_DIM    64
#define EMBED_DIM 64
#define NUM_BINS  10
#define EPS_F     1e-6f

// ---------------------------------------------------------------------------
// Kernel 1: bins = cumsum(softmax(bin_logits))
//   G[k,f] = M[k,f]-M[k+1,f] (k<9),  G[9,f] = 64*M[9,f] + 64*bias[f],
//   G[10]=G[11]=0, where M[k,f] = sum_e embed[k,e]*W[f,e].
// G is stored PAIR-INTERLEAVED for direct ds_load_b64 WMMA-B fragments:
//   wsG[(p*64 + f)*2 + o] = G[2p+o][f],  p = k>>1, o = k&1.
// ---------------------------------------------------------------------------
__global__ void k_prep(const float* __restrict__ bin_logits,
                       const float* __restrict__ embed,
                       const float* __restrict__ W,
                       const float* __restrict__ bias,
                       float* __restrict__ wsBins,   // 16 floats
                       float* __restrict__ wsG) {    // 12*64 floats (6*64 float2)
    __shared__ float sMM[NUM_BINS * 64];
    int tid = threadIdx.x;           // 768 threads: k = tid>>6 in 0..11, f = tid&63
    int k = tid >> 6, f = tid & 63;
    if (k < NUM_BINS) {
        float acc = 0.0f;
        #pragma unroll 8
        for (int e = 0; e < EMBED_DIM; ++e)
            acc += embed[k * EMBED_DIM + e] * W[f * EMBED_DIM + e];
        sMM[k * 64 + f] = acc;
    }
    __syncthreads();
    float g;
    if (k < NUM_BINS - 1)       g = sMM[k * 64 + f] - sMM[(k + 1) * 64 + f];
    else if (k == NUM_BINS - 1) g = 64.0f * sMM[(NUM_BINS - 1) * 64 + f] + 64.0f * bias[f];
    else                        g = 0.0f;   // K padding rows 10,11
    wsG[(((k >> 1) * 64) + f) * 2 + (k & 1)] = g;

    if (tid == 0) {
        float l[NUM_BINS];
        float m = -3.0e38f;
        for (int i = 0; i < NUM_BINS; ++i) { l[i] = bin_logits[i]; m = fmaxf(m, l[i]); }
        float s = 0.0f;
        for (int i = 0; i < NUM_BINS; ++i) { l[i] = expf(l[i] - m); s += l[i]; }
        float inv = 1.0f / s;
        float c = 0.0f;
        for (int i = 0; i < NUM_BINS; ++i) { c += l[i] * inv; wsBins[i] = c; }
        for (int i = NUM_BINS; i < 16; ++i) wsBins[i] = 3.0e38f;
    }
}

// ---------------------------------------------------------------------------
// Kernel 2a: per-block partial column min/max of x [B x 64] (deterministic)
// ---------------------------------------------------------------------------
__global__ void k_minmax_part(const float* __restrict__ x, int B,
                              float* __restrict__ pmin,
                              float* __restrict__ pmax) {
    __shared__ float smin[256], smax[256];
    int tid = threadIdx.x;
    int col = tid & 63, rg = tid >> 6;                 // 4 row-groups of 64 cols
    int rowsPerBlock = (B + gridDim.x - 1) / gridDim.x;
    int base = blockIdx.x * rowsPerBlock;
    int end  = base + rowsPerBlock; if (end > B) end = B;
    float vmin = 3.0e38f, vmax = -3.0e38f;
    for (int r = base + rg; r < end; r += 4) {
        float v = x[(size_t)r * 64 + col];
        vmin = fminf(vmin, v);
        vmax = fmaxf(vmax, v);
    }
    smin[tid] = vmin; smax[tid] = vmax;
    __syncthreads();
    if (rg == 0) {
        #pragma unroll
        for (int j = 1; j < 4; ++j) {
            vmin = fminf(vmin, smin[col + 64 * j]);
            vmax = fmaxf(vmax, smax[col + 64 * j]);
        }
        pmin[blockIdx.x * 64 + col] = vmin;
        pmax[blockIdx.x * 64 + col] = vmax;
    }
}

// ---------------------------------------------------------------------------
// Kernel 2b: finalize per-column normalize pair {scale, -cmin*scale}
// so that x_norm = fma(x, scale, noff)
// ---------------------------------------------------------------------------
__global__ void k_minmax_final(const float* __restrict__ pmin,
                               const float* __restrict__ pmax, int nparts,
                               float* __restrict__ wsNorm) {   // 64 x float2
    int col = threadIdx.x;  // 64 threads
    float vmin = 3.0e38f, vmax = -3.0e38f;
    for (int p = 0; p < nparts; ++p) {
        vmin = fminf(vmin, pmin[p * 64 + col]);
        vmax = fmaxf(vmax, pmax[p * 64 + col]);
    }
    float scale = 1.0f / (vmax - vmin + EPS_F);
    wsNorm[2 * col + 0] = scale;
    wsNorm[2 * col + 1] = -vmin * scale;
}

// ---------------------------------------------------------------------------
// Kernel 3: fused normalize + bucketize + (ge @ G) via v_wmma_f32_16x16x4_f32
// One wave handles 16 batch rows; 8 waves / 256-thread block.
// ---------------------------------------------------------------------------
__global__ void __launch_bounds__(256, 2)
k_main(const float* __restrict__ x,
       const float* __restrict__ wsBins,
       const float* __restrict__ wsG,
       const float* __restrict__ wsNorm,
       float* __restrict__ out, int B) {
    __shared__ float2 sGp[6 * 64];      // pair-interleaved G rows {2p, 2p+1}
    __shared__ float  sbins[16];
    __shared__ float2 snorm[64];

    int tid = threadIdx.x;
    for (int i = tid; i < 6 * 64; i += 256) sGp[i] = ((const float2*)wsG)[i];
    if (tid < 16) sbins[tid] = wsBins[tid];
    if (tid < 64) snorm[tid] = ((const float2*)wsNorm)[tid];
    __syncthreads();

    int wave = tid >> 5;
    int lane = tid & 31;
    int half = lane >> 4;          // 0: cols 0..31 & K{0,1};  1: cols 32..63 & K{2,3}
    int r    = lane & 15;          // row within tile
    int tile = blockIdx.x * 8 + wave;
    int rowbase = tile * 16;
    if (rowbase >= B) return;      // wave-uniform

    // wave-uniform bin edges -> SGPRs
    float bk[9];
    #pragma unroll
    for (int k = 0; k < 9; ++k)
        bk[k] = __int_as_float(__builtin_amdgcn_readfirstlane(__float_as_int(sbins[k])));

    // --- bucketize this lane's half row: ge[k] = #{elem : x_norm <= bins[k]} ---
    const float4* xr = (const float4*)(x + (size_t)(rowbase + r) * 64 + half * 32);
    int ge[9];
    #pragma unroll
    for (int k = 0; k < 9; ++k) ge[k] = 0;
    #pragma unroll
    for (int q = 0; q < 8; ++q) {
        float4 v4 = xr[q];
        float vv[4] = {v4.x, v4.y, v4.z, v4.w};
        #pragma unroll
        for (int j = 0; j < 4; ++j) {
            int col = half * 32 + q * 4 + j;
            float2 ns = snorm[col];
            float xn = fmaf(vv[j], ns.x, ns.y);       // (x - cmin) * scale
            #pragma unroll
            for (int k = 0; k < 9; ++k) ge[k] += (xn <= bk[k]) ? 1 : 0;
        }
    }
    // combine the two half-rows: every lane ends up with full-row ge
    #pragma unroll
    for (int k = 0; k < 9; ++k) ge[k] += __shfl_xor(ge[k], 16, 32);

    // Named scalars (NOT an array) so no dynamic-index select trees appear.
    float g0 = (float)ge[0], g1 = (float)ge[1], g2 = (float)ge[2];
    float g3 = (float)ge[3], g4 = (float)ge[4], g5 = (float)ge[5];
    float g6 = (float)ge[6], g7 = (float)ge[7], g8 = (float)ge[8];
    bool h = (half != 0);

    // A fragments (16x4 f32 layout: lanes<16 hold K0,K1; lanes>=16 hold K2,K3)
    // Padded K=12 column vector per row: [ge0..ge8, 1.0, 0, 0]
    v2f A[3];
    A[0].x = h ? g2 : g0;     A[0].y = h ? g3 : g1;
    A[1].x = h ? g6 : g4;     A[1].y = h ? g7 : g5;
    A[2].x = h ? 0.0f : g8;   A[2].y = h ? 0.0f : 1.0f;

    int n = lane & 15;
    int gbase = half * 64 + n;          // pair row p = 2c + half, runtime part
    #pragma unroll
    for (int t = 0; t < 4; ++t) {
        int n0 = 16 * t;
        v8f acc = {};                   // first WMMA takes inline-0 C operand
        #pragma unroll
        for (int c = 0; c < 3; ++c) {
            float2 p2 = sGp[gbase + c * 128 + n0];   // one ds_load_b64, adjacent pair
            v2f Bf; Bf.x = p2.x; Bf.y = p2.y;
            acc = __builtin_amdgcn_wmma_f32_16x16x4_f32(
                false, A[c], false, Bf, (short)0, acc, false, false);
        }
        // D layout: VGPR i -> row rowbase+i (lanes 0-15) / rowbase+i+8 (lanes 16-31)
        #pragma unroll
        for (int i = 0; i < 8; ++i) {
            int row = rowbase + i + 8 * half;
            out[(size_t)row * 64 + n0 + n] = acc[i];
        }
    }
}

// ---------------------------------------------------------------------------
extern "C" void kernel_launch(void* const* d_in, const int* in_sizes, int n_in,
                              void* d_out, int out_size, void* d_ws, size_t ws_size,
                              hipStream_t stream) {
    const float* x          = (const float*)d_in[0];
    const float* bin_logits = (const float*)d_in[1];
    const float* embed      = (const float*)d_in[2];
    const float* W          = (const float*)d_in[3];
    const float* b          = (const float*)d_in[4];
    float* out = (float*)d_out;
    int B = in_sizes[0] / IN_DIM;

    float* ws       = (float*)d_ws;
    float* wsBins   = ws;                // 16
    float* wsG      = ws + 16;           // 12*64 = 768 (6*64 float2, pair-interleaved)
    float* wsNorm   = ws + 800;          // 128 (64 x {scale, -cmin*scale})
    const int NPART = 128;
    float* wsPmin   = ws + 1024;                 // NPART*64
    float* wsPmax   = ws + 1024 + NPART * 64;    // NPART*64

    k_prep<<<1, 768, 0, stream>>>(bin_logits, embed, W, b, wsBins, wsG);
    k_minmax_part<<<NPART, 256, 0, stream>>>(x, B, wsPmin, wsPmax);
    k_minmax_final<<<1, 64, 0, stream>>>(wsPmin, wsPmax, NPART, wsNorm);

    int tiles  = (B + 15) / 16;
    int blocks = (tiles + 7) / 8;
    k_main<<<blocks, 256, 0, stream>>>(x, wsBins, wsG, wsNorm, out, B);
}